// Encoder_46935402611143
// MI455X (gfx1250) — compile-verified
//
#include <hip/hip_runtime.h>
#include <hip/hip_bf16.h>
#include <math.h>

#define H_  2
#define C_  32
#define D_  64
#define EC_ 5
#define IN_ 4
#define L_  4

typedef float v2f __attribute__((ext_vector_type(2)));
typedef float v8f __attribute__((ext_vector_type(8)));

#if defined(__HIP_DEVICE_COMPILE__) && defined(__gfx1250__) && __has_builtin(__builtin_amdgcn_wmma_f32_16x16x4_f32)
#define HAVE_WMMA_F32X4 1
#else
#define HAVE_WMMA_F32X4 0
#endif

// ---- monotonic float <-> uint key for atomicMax-based segment max ----
__device__ __forceinline__ unsigned f32_key(float f) {
  unsigned u = __float_as_uint(f);
  return (u & 0x80000000u) ? ~u : (u | 0x80000000u);
}
__device__ __forceinline__ float key_f32(unsigned k) {
  unsigned u = (k & 0x80000000u) ? (k & 0x7FFFFFFFu) : ~k;
  return __uint_as_float(u);
}
#define NEG_INF_KEY 0x007FFFFFu  // f32_key(-inf)

// ================= layer 0: hp = emb_table[x] @ W0  (N x 4 @ 4 x 64) ==========
__global__ void layer0_kernel(const int* __restrict__ x, const float* __restrict__ emb,
                              const float* __restrict__ W0, float* __restrict__ hp, int n) {
  int t = blockIdx.x * blockDim.x + threadIdx.x;
  if (t >= n * D_) return;
  int node = t >> 6, c = t & 63;
  int ty = x[node];
  const float* e = emb + ty * IN_;
  float acc = 0.f;
#pragma unroll
  for (int k = 0; k < IN_; k++) acc = fmaf(e[k], W0[k * D_ + c], acc);
  hp[t] = acc;
}

// ================= dense GEMM: hp = h @ W  (N x 64 @ 64 x 64), fp32 WMMA ======
// one wave per 16x16 output tile; K loop 64 in steps of 4 (V_WMMA_F32_16X16X4_F32)
__global__ void gemm64_wmma(const float* __restrict__ h, const float* __restrict__ W,
                            float* __restrict__ hp, int nrows) {
  int wavesPerBlock = blockDim.x >> 5;
  int wave = blockIdx.x * wavesPerBlock + (threadIdx.x >> 5);
  int lane = threadIdx.x & 31;
  int mtiles = (nrows + 15) >> 4;
  int mtile = wave >> 2;   // 4 n-tiles (of 16) across D_=64
  int ntile = wave & 3;
  if (mtile >= mtiles) return;
  int m0 = mtile << 4, n0 = ntile << 4;
  int lr = lane & 15;          // row-in-tile (A) / col-in-tile (B, C/D)
  int hk = (lane >> 4) << 1;   // K sub-offset: lanes 0-15 -> {0,1}, 16-31 -> {2,3}

  int arow = m0 + lr; if (arow >= nrows) arow = nrows - 1;
  const float* ap = h + (size_t)arow * D_;

  v8f acc = {};
#if HAVE_WMMA_F32X4
#pragma unroll
  for (int k0 = 0; k0 < D_; k0 += 4) {
    v2f a, b;
    a.x = ap[k0 + hk];
    a.y = ap[k0 + hk + 1];
    b.x = W[(k0 + hk) * D_ + n0 + lr];
    b.y = W[(k0 + hk + 1) * D_ + n0 + lr];
    acc = __builtin_amdgcn_wmma_f32_16x16x4_f32(false, a, false, b, (short)0, acc,
                                                false, false);
  }
#else
  // scalar fallback: each lane computes its 8 C/D elements directly
  {
    int nn = n0 + lr;
    int mbase = m0 + ((lane >> 4) << 3);
#pragma unroll
    for (int r = 0; r < 8; r++) {
      int row = mbase + r; if (row >= nrows) row = nrows - 1;
      float s = 0.f;
      for (int k = 0; k < D_; k++) s = fmaf(h[(size_t)row * D_ + k], W[k * D_ + nn], s);
      acc[r] = s;
    }
  }
#endif
  // C/D layout: VGPR r -> (M = r + 8*(lane>=16), N = lane%16)
  int mbase = m0 + ((lane >> 4) << 3);
#pragma unroll
  for (int r = 0; r < 8; r++) {
    int row = mbase + r;
    if (row < nrows) hp[(size_t)row * D_ + n0 + lr] = acc[r];
  }
}

// ============ per-node attention dots: s_src/s_dst [N,H] ======================
__global__ void node_dots(const float* __restrict__ hp, const float* __restrict__ a_s,
                          const float* __restrict__ a_d, float* __restrict__ ssrc,
                          float* __restrict__ sdst, int n) {
  int t = blockIdx.x * blockDim.x + threadIdx.x;
  if (t >= n * H_) return;
  int node = t >> 1, head = t & 1;
  const float* hrow = hp + (size_t)node * D_ + head * C_;
  const float* as = a_s + head * C_;
  const float* ad = a_d + head * C_;
  float s1 = 0.f, s2 = 0.f;
#pragma unroll
  for (int c = 0; c < C_; c++) {
    float v = hrow[c];
    s1 = fmaf(v, as[c], s1);
    s2 = fmaf(v, ad[c], s2);
  }
  ssrc[t] = s1;
  sdst[t] = s2;
}

// ============ per-edge-class attention dots: eatt [EC,H] ======================
__global__ void eatt_kernel(const float* __restrict__ ee, const float* __restrict__ a_s,
                            float* __restrict__ eatt) {
  int t = threadIdx.x;
  if (t >= EC_ * H_) return;
  int ec = t >> 1, head = t & 1;
  const float* er = ee + ec * D_ + head * C_;
  const float* as = a_s + head * C_;
  float s = 0.f;
#pragma unroll
  for (int c = 0; c < C_; c++) s = fmaf(er[c], as[c], s);
  eatt[t] = s;
}

// ============ init mxkey / den ================================================
__global__ void init_nodeh(unsigned* __restrict__ mxkey, float* __restrict__ den, int nh) {
  int t = blockIdx.x * blockDim.x + threadIdx.x;
  if (t >= nh) return;
  mxkey[t] = NEG_INF_KEY;
  den[t] = 0.f;
}

// ============ edge logits + atomic segment max ================================
__global__ void edge_logits(const int* __restrict__ src, const int* __restrict__ dst,
                            const int* __restrict__ ea, const float* __restrict__ ssrc,
                            const float* __restrict__ sdst, const float* __restrict__ eatt,
                            float* __restrict__ logits, unsigned* __restrict__ mxkey, int E) {
  int t = blockIdx.x * blockDim.x + threadIdx.x;
  if (t >= E * H_) return;
  int e = t >> 1, head = t & 1;
  int s = src[e], d = dst[e], c = ea[e];
  float lg = ssrc[s * H_ + head] + eatt[c * H_ + head] + sdst[d * H_ + head];
  lg = (lg > 0.f) ? lg : 0.2f * lg;      // leaky_relu(0.2)
  logits[t] = lg;
  atomicMax(&mxkey[d * H_ + head], f32_key(lg));
}

// ============ decode segment max (non-finite -> 0, matching reference) ========
__global__ void mx_decode(const unsigned* __restrict__ mxkey, float* __restrict__ mxf, int nh) {
  int t = blockIdx.x * blockDim.x + threadIdx.x;
  if (t >= nh) return;
  float m = key_f32(mxkey[t]);
  mxf[t] = isfinite(m) ? m : 0.f;
}

// ============ exp + atomic segment sum ========================================
__global__ void edge_exp(const int* __restrict__ dst, float* __restrict__ logits,
                         const float* __restrict__ mxf, float* __restrict__ den, int E) {
  int t = blockIdx.x * blockDim.x + threadIdx.x;
  if (t >= E * H_) return;
  int e = t >> 1, head = t & 1;
  int d = dst[e];
  float ev = expf(logits[t] - mxf[d * H_ + head]);
  logits[t] = ev;                        // reuse as numerator
  atomicAdd(&den[d * H_ + head], ev);
}

// ============ weighted message aggregation (E x 64 threads) ===================
__global__ void aggregate(const int* __restrict__ src, const int* __restrict__ dst,
                          const int* __restrict__ ea, const float* __restrict__ num,
                          const float* __restrict__ den, const float* __restrict__ hp,
                          const float* __restrict__ ee, float* __restrict__ out, int E) {
  int t = blockIdx.x * blockDim.x + threadIdx.x;
  if (t >= E * D_) return;
  int e = t >> 6, c = t & 63;
  int head = c >> 5;
  int d = dst[e], s = src[e], cls = ea[e];
  float alpha = num[e * H_ + head] / (den[d * H_ + head] + 1e-16f);
  float m = hp[(size_t)s * D_ + c] + ee[cls * D_ + c];
  atomicAdd(&out[(size_t)d * D_ + c], alpha * m);
}

// ============ elementwise ELU (in place) ======================================
__global__ void elu_kernel(float* __restrict__ v, int n) {
  int t = blockIdx.x * blockDim.x + threadIdx.x;
  if (t >= n) return;
  float x = v[t];
  v[t] = (x > 0.f) ? x : expm1f(x);
}

// ============ user/item split: count / scan / scatter =========================
__global__ void count_zeros(const int* __restrict__ x, int* __restrict__ blockCnt, int n) {
  __shared__ int s[256];
  int node = blockIdx.x * 256 + threadIdx.x;
  s[threadIdx.x] = (node < n && x[node] == 0) ? 1 : 0;
  __syncthreads();
  for (int off = 128; off > 0; off >>= 1) {
    if (threadIdx.x < off) s[threadIdx.x] += s[threadIdx.x + off];
    __syncthreads();
  }
  if (threadIdx.x == 0) blockCnt[blockIdx.x] = s[0];
}

__global__ void scan_blocks(const int* __restrict__ blockCnt, int* __restrict__ blockOff, int nb) {
  if (blockIdx.x == 0 && threadIdx.x == 0) {
    int acc = 0;
    for (int i = 0; i < nb; i++) { blockOff[i] = acc; acc += blockCnt[i]; }
    blockOff[nb] = acc;   // total zeros
  }
}

__global__ void scatter_out(const float* __restrict__ h, const int* __restrict__ x,
                            const int* __restrict__ blockOff, float* __restrict__ out,
                            int n, int nb) {
  __shared__ int sflag[256];
  int tid = threadIdx.x;
  int node = blockIdx.x * 256 + tid;
  int f = (node < n && x[node] == 0) ? 1 : 0;
  sflag[tid] = f;
  __syncthreads();
  // Hillis-Steele inclusive scan over 256 flags
  for (int off = 1; off < 256; off <<= 1) {
    int v = (tid >= off) ? sflag[tid - off] : 0;
    __syncthreads();
    sflag[tid] += v;
    __syncthreads();
  }
  if (node >= n) return;
  int zerosIncl = blockOff[blockIdx.x] + sflag[tid];
  int zerosBefore = zerosIncl - f;
  int total0 = blockOff[nb];
  long dstRow = f ? (long)zerosBefore : (long)total0 + (node - zerosBefore);
  const float4* sv = (const float4*)(h + (size_t)node * D_);
  float4* dv = (float4*)(out + (size_t)dstRow * D_);
#pragma unroll
  for (int i = 0; i < D_ / 4; i++) dv[i] = sv[i];
}

// =============================================================================
extern "C" void kernel_launch(void* const* d_in, const int* in_sizes, int n_in,
                              void* d_out, int out_size, void* d_ws, size_t ws_size,
                              hipStream_t stream) {
  (void)n_in; (void)out_size; (void)ws_size;
  const int*   x       = (const int*)d_in[0];
  const int*   eidx    = (const int*)d_in[1];
  const int*   eattr   = (const int*)d_in[2];
  const float* emb     = (const float*)d_in[3];
  const float* W0      = (const float*)d_in[4];
  const float* Ws      = (const float*)d_in[5];
  const float* att_src = (const float*)d_in[6];
  const float* att_dst = (const float*)d_in[7];
  const float* eembs   = (const float*)d_in[8];
  float* out = (float*)d_out;

  const int N = in_sizes[0];
  const int E = in_sizes[1] / 2;
  const int* src = eidx;
  const int* dst = eidx + E;

  // ---- workspace carve-up (256B aligned) ----
  char* w = (char*)d_ws;
  size_t off = 0;
  auto carve = [&](size_t bytes) -> void* {
    void* p = w + off;
    off = (off + bytes + 255) & ~(size_t)255;
    return p;
  };
  float* buf[3];
  buf[0] = (float*)carve((size_t)N * D_ * sizeof(float));
  buf[1] = (float*)carve((size_t)N * D_ * sizeof(float));
  buf[2] = (float*)carve((size_t)N * D_ * sizeof(float));
  float*    ssrc   = (float*)carve((size_t)N * H_ * sizeof(float));
  float*    sdst   = (float*)carve((size_t)N * H_ * sizeof(float));
  unsigned* mxkey  = (unsigned*)carve((size_t)N * H_ * sizeof(unsigned));
  float*    mxf    = (float*)carve((size_t)N * H_ * sizeof(float));
  float*    den    = (float*)carve((size_t)N * H_ * sizeof(float));
  float*    eatt   = (float*)carve(EC_ * H_ * sizeof(float));
  float*    logits = (float*)carve((size_t)E * H_ * sizeof(float));
  int nb = (N + 255) / 256;
  int* blockCnt = (int*)carve((size_t)nb * sizeof(int));
  int* blockOff = (int*)carve((size_t)(nb + 1) * sizeof(int));

  const int nh = N * H_;
  const int eh = E * H_;
  const int nd = N * D_;
  const int ed = E * D_;

  auto run_attention = [&](int l, const float* hp, float* acc) {
    const float* a_s = att_src + l * H_ * C_;
    const float* a_d = att_dst + l * H_ * C_;
    const float* ee  = eembs + l * EC_ * D_;
    node_dots<<<(nh + 255) / 256, 256, 0, stream>>>(hp, a_s, a_d, ssrc, sdst, N);
    eatt_kernel<<<1, 32, 0, stream>>>(ee, a_s, eatt);
    init_nodeh<<<(nh + 255) / 256, 256, 0, stream>>>(mxkey, den, nh);
    edge_logits<<<(eh + 255) / 256, 256, 0, stream>>>(src, dst, eattr, ssrc, sdst, eatt,
                                                      logits, mxkey, E);
    mx_decode<<<(nh + 255) / 256, 256, 0, stream>>>(mxkey, mxf, nh);
    edge_exp<<<(eh + 255) / 256, 256, 0, stream>>>(dst, logits, mxf, den, E);
    hipMemsetAsync(acc, 0, (size_t)N * D_ * sizeof(float), stream);
    aggregate<<<(ed + 255) / 256, 256, 0, stream>>>(src, dst, eattr, logits, den, hp, ee,
                                                    acc, E);
    elu_kernel<<<(nd + 255) / 256, 256, 0, stream>>>(acc, nd);
  };

  // ---- layer 0: embedding lookup + tiny GEMM (IN=4) ----
  layer0_kernel<<<(nd + 255) / 256, 256, 0, stream>>>(x, emb, W0, buf[0], N);
  run_attention(0, buf[0], buf[1]);
  int hcur = 1;

  // ---- layers 1..3: WMMA fp32 GEMM + attention ----
  int mtiles = (N + 15) / 16;
  int waves = mtiles * (D_ / 16);
  int gemmBlocks = (waves + 7) / 8;   // 8 waves (256 threads) per block
  for (int l = 1; l < L_; l++) {
    int hp = (hcur + 1) % 3;
    int ac = (hcur + 2) % 3;
    gemm64_wmma<<<gemmBlocks, 256, 0, stream>>>(buf[hcur], Ws + (size_t)(l - 1) * D_ * D_,
                                                buf[hp], N);
    run_attention(l, buf[hp], buf[ac]);
    hcur = ac;
  }

  // ---- split into (users, items) preserving index order ----
  count_zeros<<<nb, 256, 0, stream>>>(x, blockCnt, N);
  scan_blocks<<<1, 32, 0, stream>>>(blockCnt, blockOff, nb);
  scatter_out<<<nb, 256, 0, stream>>>(buf[hcur], x, blockOff, out, N, nb);
}